// EdgeDegreeEmbeddingNetwork_61040075211258
// MI455X (gfx1250) — compile-verified
//
#include <hip/hip_runtime.h>

typedef float v2f __attribute__((ext_vector_type(2)));
typedef float v8f __attribute__((ext_vector_type(8)));

#define H_      128
#define N_      10000
#define E_      160000
#define RAD_IN_ 64
#define RAD_HID_ 64
#define WNUM_   384
#define MUL0_   128
#define MUL1_   64
#define MUL2_   32
#define OUTD_   480   // 128 + 3*64 + 5*32

// Native CDNA5 fire-and-forget f32 atomic add (no return -> STOREcnt only,
// guaranteed GLOBAL_ATOMIC_ADD_F32, never a CAS loop).
__device__ __forceinline__ void gatomic_fadd(float* p, float v) {
  asm volatile("global_atomic_add_f32 %0, %1, off" : : "v"(p), "v"(v) : "memory");
}

// ---------------------------------------------------------------------------
// node_feat[n,h] = ring_emb[ring_info[n], h] + sum_k onehot[n,k] * atom_W[h,k]
// ---------------------------------------------------------------------------
__global__ void node_feat_kernel(const float* __restrict__ onehot,
                                 const float* __restrict__ ring_emb,
                                 const float* __restrict__ atom_W,
                                 const int*   __restrict__ ring_info,
                                 float*       __restrict__ nf) {
  int idx = blockIdx.x * blockDim.x + threadIdx.x;
  if (idx >= N_ * H_) return;
  int n = idx >> 7, h = idx & 127;
  float acc = ring_emb[ring_info[n] * H_ + h];
  const float* oh = onehot + n * 16;
  const float* aw = atom_W + h * 16;
#pragma unroll
  for (int k = 0; k < 16; ++k) acc += oh[k] * aw[k];
  nf[idx] = acc;
}

__global__ void zero_kernel(float* __restrict__ p, int n) {
  int idx = blockIdx.x * blockDim.x + threadIdx.x;
  if (idx < n) p[idx] = 0.f;
}

// ---------------------------------------------------------------------------
// Edge pipeline: one wave32 per 16-edge tile, 2 waves per block.
// All GEMMs on V_WMMA_F32_16X16X4_F32 (full fp32 precision on matrix pipes).
// ---------------------------------------------------------------------------
__launch_bounds__(64)
__global__ void edge_kernel(const float* __restrict__ edge_attr,
                            const float* __restrict__ edge_scalars,
                            const float* __restrict__ rad_W1,
                            const float* __restrict__ rad_b1,
                            const float* __restrict__ rad_g,
                            const float* __restrict__ rad_b,
                            const float* __restrict__ rad_W2,
                            const float* __restrict__ rad_offset,
                            const float* __restrict__ proj_W0,
                            const float* __restrict__ proj_b0,
                            const float* __restrict__ proj_W1,
                            const float* __restrict__ proj_W2,
                            const int*   __restrict__ edge_src,
                            const int*   __restrict__ edge_dst,
                            const float* __restrict__ nf,
                            float*       __restrict__ out) {
  __shared__ float sh_h[2][16][RAD_HID_];  // post-LN/SiLU radial hidden
  __shared__ float sh_w[2][16][WNUM_];     // radial weights w0|w1|w2 per edge

  const int lane = threadIdx.x & 31;
  const int wv   = threadIdx.x >> 5;       // wave in block (0..1)
  const int half = lane >> 4;              // lane-half selects K/M offset
  const int l15  = lane & 15;
  const int base = (blockIdx.x * 2 + wv) * 16;  // first edge of this tile

  // ---- GEMM1: S1 = edge_scalars(16x64) @ rad_W1^T (64x64) + b1 -> LDS ----
  // A fragment: row=l15 edge, K pair at 2*half. B fragment: B[k][n]=W1[n][k].
  {
    const float* aRow = edge_scalars + (size_t)(base + l15) * RAD_IN_ + 2 * half;
    for (int nt = 0; nt < RAD_HID_ / 16; ++nt) {
      v8f acc = {};
      const float* bRow = rad_W1 + (size_t)(nt * 16 + l15) * RAD_IN_ + 2 * half;
      for (int kt = 0; kt < RAD_IN_ / 4; ++kt) {
        v2f a = *(const v2f*)(aRow + kt * 4);
        v2f b = *(const v2f*)(bRow + kt * 4);
        acc = __builtin_amdgcn_wmma_f32_16x16x4_f32(false, a, false, b,
                                                    (short)0, acc, false, false);
      }
      const int col = nt * 16 + l15;
      const float b1 = rad_b1[col];
#pragma unroll
      for (int v = 0; v < 8; ++v)
        sh_h[wv][v + half * 8][col] = acc[v] + b1;
    }
  }
  __syncthreads();

  // ---- LayerNorm + SiLU: row = l15, each lane-half owns 32 columns; ----
  // ---- partial mean/var combined across halves with one shfl_xor(16). ----
  {
    float* rowp = sh_h[wv][l15];
    const int c0 = half * 32;
    float s = 0.f;
    for (int c = 0; c < 32; ++c) s += rowp[c0 + c];
    s += __shfl_xor(s, 16, 32);
    const float mu = s * (1.f / RAD_HID_);
    float var = 0.f;
    for (int c = 0; c < 32; ++c) { float d = rowp[c0 + c] - mu; var += d * d; }
    var += __shfl_xor(var, 16, 32);
    const float rs = rsqrtf(var * (1.f / RAD_HID_) + 1e-5f);
    for (int c = c0; c < c0 + 32; ++c) {
      float t = (rowp[c] - mu) * rs * rad_g[c] + rad_b[c];
      rowp[c] = t / (1.f + __expf(-t));  // silu
    }
  }
  __syncthreads();

  // ---- GEMM2: W = h(16x64) @ rad_W2^T (64x384) + offset -> LDS ----
  {
    const float* aRow = &sh_h[wv][l15][2 * half];
    for (int nt = 0; nt < WNUM_ / 16; ++nt) {
      v8f acc = {};
      const float* bRow = rad_W2 + (size_t)(nt * 16 + l15) * RAD_HID_ + 2 * half;
      for (int kt = 0; kt < RAD_HID_ / 4; ++kt) {
        v2f a = *(const v2f*)(aRow + kt * 4);
        v2f b = *(const v2f*)(bRow + kt * 4);
        acc = __builtin_amdgcn_wmma_f32_16x16x4_f32(false, a, false, b,
                                                    (short)0, acc, false, false);
      }
      const int col = nt * 16 + l15;
      const float off = rad_offset[col];
#pragma unroll
      for (int v = 0; v < 8; ++v)
        sh_w[wv][v + half * 8][col] = acc[v] + off;
    }
  }
  __syncthreads();

  // Gathered node-feature row for this lane's edge (L2-hot table).
  const float* xrow = nf + (size_t)edge_src[base + l15] * H_;
  const float* wrow = sh_w[wv][l15];

  const float k0s = 0.08838834764831845f * 0.25f;   // 1/sqrt(128) * 1/sqrt(16)
  const float k1s = 0.05103103630798288f * 0.25f;   // /sqrt(3)
  const float k2s = 0.03952847075210474f * 0.25f;   // /sqrt(5)

  // ---- GEMM3: p0 = (x*w0)(16x128) @ proj_W0^T (128x128), scatter-add ----
  for (int nt = 0; nt < MUL0_ / 16; ++nt) {
    v8f acc = {};
    const float* bRow = proj_W0 + (size_t)(nt * 16 + l15) * H_ + 2 * half;
    for (int kt = 0; kt < H_ / 4; ++kt) {
      const int kk = kt * 4 + 2 * half;
      v2f a; a.x = xrow[kk] * wrow[kk]; a.y = xrow[kk + 1] * wrow[kk + 1];
      v2f b = *(const v2f*)(bRow + kt * 4);
      acc = __builtin_amdgcn_wmma_f32_16x16x4_f32(false, a, false, b,
                                                  (short)0, acc, false, false);
    }
    const int col = nt * 16 + l15;
    const float bias = proj_b0[col];
#pragma unroll
    for (int v = 0; v < 8; ++v) {
      const int e = base + v + half * 8;
      const float y0 = edge_attr[e * 9];
      const float val = acc[v] * y0 * k0s + bias * 0.25f;
      gatomic_fadd(out + (size_t)edge_dst[e] * OUTD_ + col, val);
    }
  }

  // ---- GEMM4: p1 = (x*w1)(16x128) @ proj_W1^T (128x64), x3 outer -> cols 128.. ----
  for (int nt = 0; nt < MUL1_ / 16; ++nt) {
    v8f acc = {};
    const float* bRow = proj_W1 + (size_t)(nt * 16 + l15) * H_ + 2 * half;
    for (int kt = 0; kt < H_ / 4; ++kt) {
      const int kk = kt * 4 + 2 * half;
      v2f a; a.x = xrow[kk] * wrow[MUL0_ + kk]; a.y = xrow[kk + 1] * wrow[MUL0_ + kk + 1];
      v2f b = *(const v2f*)(bRow + kt * 4);
      acc = __builtin_amdgcn_wmma_f32_16x16x4_f32(false, a, false, b,
                                                  (short)0, acc, false, false);
    }
    const int mul = nt * 16 + l15;
#pragma unroll
    for (int v = 0; v < 8; ++v) {
      const int e = base + v + half * 8;
      float* orow = out + (size_t)edge_dst[e] * OUTD_ + MUL0_ + mul * 3;
      const float bv = acc[v] * k1s;
#pragma unroll
      for (int j = 0; j < 3; ++j)
        gatomic_fadd(orow + j, bv * edge_attr[e * 9 + 1 + j]);
    }
  }

  // ---- GEMM5: p2 = (x*w2)(16x128) @ proj_W2^T (128x32), x5 outer -> cols 320.. ----
  for (int nt = 0; nt < MUL2_ / 16; ++nt) {
    v8f acc = {};
    const float* bRow = proj_W2 + (size_t)(nt * 16 + l15) * H_ + 2 * half;
    for (int kt = 0; kt < H_ / 4; ++kt) {
      const int kk = kt * 4 + 2 * half;
      v2f a; a.x = xrow[kk] * wrow[2 * MUL0_ + kk]; a.y = xrow[kk + 1] * wrow[2 * MUL0_ + kk + 1];
      v2f b = *(const v2f*)(bRow + kt * 4);
      acc = __builtin_amdgcn_wmma_f32_16x16x4_f32(false, a, false, b,
                                                  (short)0, acc, false, false);
    }
    const int mul = nt * 16 + l15;
#pragma unroll
    for (int v = 0; v < 8; ++v) {
      const int e = base + v + half * 8;
      float* orow = out + (size_t)edge_dst[e] * OUTD_ + MUL0_ + 3 * MUL1_ + mul * 5;
      const float bv = acc[v] * k2s;
#pragma unroll
      for (int j = 0; j < 5; ++j)
        gatomic_fadd(orow + j, bv * edge_attr[e * 9 + 4 + j]);
    }
  }
}

// ---------------------------------------------------------------------------
extern "C" void kernel_launch(void* const* d_in, const int* in_sizes, int n_in,
                              void* d_out, int out_size, void* d_ws, size_t ws_size,
                              hipStream_t stream) {
  const float* onehot       = (const float*)d_in[0];
  const float* edge_attr    = (const float*)d_in[1];
  const float* edge_scalars = (const float*)d_in[2];
  const float* ring_emb     = (const float*)d_in[3];
  const float* atom_W       = (const float*)d_in[4];
  const float* rad_W1       = (const float*)d_in[5];
  const float* rad_b1       = (const float*)d_in[6];
  const float* rad_g        = (const float*)d_in[7];
  const float* rad_b        = (const float*)d_in[8];
  const float* rad_W2       = (const float*)d_in[9];
  const float* rad_offset   = (const float*)d_in[10];
  const float* proj_W0      = (const float*)d_in[11];
  const float* proj_b0      = (const float*)d_in[12];
  const float* proj_W1      = (const float*)d_in[13];
  const float* proj_W2      = (const float*)d_in[14];
  const int*   ring_info    = (const int*)d_in[15];
  const int*   edge_src     = (const int*)d_in[16];
  const int*   edge_dst     = (const int*)d_in[17];

  float* out = (float*)d_out;
  float* nf  = (float*)d_ws;  // N_*H_ floats of scratch for node features

  zero_kernel<<<(N_ * OUTD_ + 255) / 256, 256, 0, stream>>>(out, N_ * OUTD_);
  node_feat_kernel<<<(N_ * H_ + 255) / 256, 256, 0, stream>>>(
      onehot, ring_emb, atom_W, ring_info, nf);
  edge_kernel<<<E_ / 32, 64, 0, stream>>>(
      edge_attr, edge_scalars, rad_W1, rad_b1, rad_g, rad_b, rad_W2, rad_offset,
      proj_W0, proj_b0, proj_W1, proj_W2, edge_src, edge_dst, nf, out);
}